// PyGGCN_87230785782111
// MI455X (gfx1250) — compile-verified
//
#include <hip/hip_runtime.h>
#include <hip/hip_bf16.h>
#include <math.h>

#define NODES 100000
#define EDGES 3200000
#define FIN   512
#define FHID  16
#define FOUT  40

typedef __attribute__((ext_vector_type(2))) float v2f;
typedef __attribute__((ext_vector_type(4))) float v4f;
typedef __attribute__((ext_vector_type(8))) float v8f;

// ---------------------------------------------------------------- utilities
__global__ void fill_f32(float* __restrict__ p, float v, int n) {
    int i = blockIdx.x * blockDim.x + threadIdx.x;
    if (i < n) p[i] = v;
}

__global__ void deg_count_kernel(const long long* __restrict__ dst,
                                 float* __restrict__ deg, int e) {
    int i = blockIdx.x * blockDim.x + threadIdx.x;
    if (i < e) unsafeAtomicAdd(&deg[(int)dst[i]], 1.0f);
}

__global__ void dinv_kernel(float* __restrict__ deg, int n) {
    int i = blockIdx.x * blockDim.x + threadIdx.x;
    if (i < n) {
        float d = deg[i];
        deg[i] = (d > 0.0f) ? rsqrtf(d) : 0.0f;
    }
}

// ------------------------------------------------- GEMM1: h1 = x @ W1 (f32 WMMA)
// x:[N,512] row-major (streamed once -> non-temporal), W1:[512,16], h1:[N,16].
// One wave per 16-row tile; V_WMMA_F32_16X16X4_F32, 128 k-steps.
// A (16x4 f32): lane<16 -> M=lane, K=k0+{0,1}; lane>=16 -> M=lane-16, K=k0+{2,3}.
// B (4x16 f32): lane half selects K pair, N = lane&15.
// C: VGPR r -> M = r + 8*(lane>=16), N = lane&15.
__global__ void gemm1_wmma_kernel(const float* __restrict__ x,
                                  const float* __restrict__ W1,
                                  float* __restrict__ h1) {
    const int tiles = NODES / 16;                          // 6250 (exact)
    int wave = blockIdx.x * (blockDim.x >> 5) + (threadIdx.x >> 5);
    if (wave >= tiles) return;                             // wave-uniform guard
    int lane = threadIdx.x & 31;
    int half = lane >> 4;                                  // 0 | 1
    int l16  = lane & 15;
    int koff = half << 1;                                  // 0 | 2
    // x row base is 2048B-aligned; ka is even -> 8B-aligned v2f loads.
    const float* xrow = x + (size_t)(wave * 16 + l16) * FIN;
    v8f c = {};
    for (int k0 = 0; k0 < FIN; k0 += 4) {
        int ka = k0 + koff;
        // non-temporal: x is a 205MB one-shot stream; keep it out of the way
        // of the L2-resident edge-index / h1 working set.
        v2f a = __builtin_nontemporal_load((const v2f*)(xrow + ka));
        v2f b;
        b.x = W1[ka * FHID + l16];
        b.y = W1[(ka + 1) * FHID + l16];
        c = __builtin_amdgcn_wmma_f32_16x16x4_f32(false, a, false, b,
                                                  (short)0, c, false, false);
    }
    float* o = h1 + (size_t)(wave * 16 + (half << 3)) * FHID + l16;
#pragma unroll
    for (int r = 0; r < 8; ++r) o[(size_t)r * FHID] = c[r];
}

// ------------------------------- GEMM2: h2 = relu(agg1 + b1) @ W2 (f32 WMMA)
// agg1:[N,16], W2:[16,40] row-major, h2:[N,40]. N padded 40 -> 48 (3 N-tiles),
// third tile data-masked with zeros (no EXEC divergence around WMMA).
__global__ void gemm2_wmma_kernel(const float* __restrict__ agg1,
                                  const float* __restrict__ b1,
                                  const float* __restrict__ W2,
                                  float* __restrict__ h2) {
    const int tiles = NODES / 16;
    int wave = blockIdx.x * (blockDim.x >> 5) + (threadIdx.x >> 5);
    if (wave >= tiles) return;
    int lane = threadIdx.x & 31;
    int half = lane >> 4;
    int l16  = lane & 15;
    int koff = half << 1;
    const float* arow = agg1 + (size_t)(wave * 16 + l16) * FHID;
    v8f c0 = {}, c1 = {}, c2 = {};
#pragma unroll
    for (int k0 = 0; k0 < FHID; k0 += 4) {
        int ka = k0 + koff;
        v2f a;
        a.x = fmaxf(arow[ka]     + b1[ka],     0.0f);      // fused bias + ReLU
        a.y = fmaxf(arow[ka + 1] + b1[ka + 1], 0.0f);
        v2f bb0, bb1, bb2;
        bb0.x = W2[ka * FOUT + l16];
        bb0.y = W2[(ka + 1) * FOUT + l16];
        bb1.x = W2[ka * FOUT + 16 + l16];
        bb1.y = W2[(ka + 1) * FOUT + 16 + l16];
        int c32 = 32 + l16;
        bb2.x = (c32 < FOUT) ? W2[ka * FOUT + c32]       : 0.0f;
        bb2.y = (c32 < FOUT) ? W2[(ka + 1) * FOUT + c32] : 0.0f;
        c0 = __builtin_amdgcn_wmma_f32_16x16x4_f32(false, a, false, bb0,
                                                   (short)0, c0, false, false);
        c1 = __builtin_amdgcn_wmma_f32_16x16x4_f32(false, a, false, bb1,
                                                   (short)0, c1, false, false);
        c2 = __builtin_amdgcn_wmma_f32_16x16x4_f32(false, a, false, bb2,
                                                   (short)0, c2, false, false);
    }
    float* o = h2 + (size_t)(wave * 16 + (half << 3)) * FOUT;
#pragma unroll
    for (int r = 0; r < 8; ++r) {
        o[(size_t)r * FOUT + l16]      = c0[r];
        o[(size_t)r * FOUT + 16 + l16] = c1[r];
        if (l16 < FOUT - 32) o[(size_t)r * FOUT + 32 + l16] = c2[r];
    }
}

// ----------------------------------------- edge gather-scale-scatter (atomics)
// Work items 0..E-1: real edges; E..E+N-1: self-loops (s == d == t-E).
// F is a multiple of 4 and all row bases are 16B-aligned -> b128 gathers.
template <int F>
__global__ void edge_agg_kernel(const long long* __restrict__ src,
                                const long long* __restrict__ dst,
                                const float* __restrict__ dinv,
                                const float* __restrict__ h,
                                float* __restrict__ agg) {
    long long t = (long long)blockIdx.x * blockDim.x + threadIdx.x;
    if (t >= (long long)EDGES + NODES) return;
    int s, d;
    if (t < EDGES) {
        s = (int)src[t];
        d = (int)dst[t];
    } else {
        s = d = (int)(t - EDGES);
    }
    float nrm = dinv[s] * dinv[d];
    const v4f* hs = (const v4f*)(h + (size_t)s * F);       // global_load_b128
    float* ad = agg + (size_t)d * F;
#pragma unroll
    for (int q = 0; q < F / 4; ++q) {
        v4f m = hs[q];
        unsafeAtomicAdd(&ad[q * 4 + 0], m.x * nrm);
        unsafeAtomicAdd(&ad[q * 4 + 1], m.y * nrm);
        unsafeAtomicAdd(&ad[q * 4 + 2], m.z * nrm);
        unsafeAtomicAdd(&ad[q * 4 + 3], m.w * nrm);
    }
}

// ---------------------------------------- out = log_softmax(agg2 + b2), in place
__global__ void finalize_kernel(float* __restrict__ out,
                                const float* __restrict__ b2, int n) {
    int i = blockIdx.x * blockDim.x + threadIdx.x;
    if (i >= n) return;
    float* row = out + (size_t)i * FOUT;
    float v[FOUT];
    float m = -3.402823466e38f;
#pragma unroll
    for (int f = 0; f < FOUT; ++f) { v[f] = row[f] + b2[f]; m = fmaxf(m, v[f]); }
    float s = 0.0f;
#pragma unroll
    for (int f = 0; f < FOUT; ++f) s += __expf(v[f] - m);
    float ls = __logf(s);
#pragma unroll
    for (int f = 0; f < FOUT; ++f) row[f] = v[f] - m - ls;
}

// ---------------------------------------------------------------- launcher
extern "C" void kernel_launch(void* const* d_in, const int* in_sizes, int n_in,
                              void* d_out, int out_size, void* d_ws, size_t ws_size,
                              hipStream_t stream) {
    const float*     x    = (const float*)d_in[0];        // [N,512]
    const long long* ei   = (const long long*)d_in[1];    // [2,E] int64
    const float*     W1   = (const float*)d_in[2];        // [512,16]
    const float*     b1   = (const float*)d_in[3];        // [16]
    const float*     W2   = (const float*)d_in[4];        // [16,40]
    const float*     b2   = (const float*)d_in[5];        // [40]
    const long long* srcI = ei;                           // row 0
    const long long* dstI = ei + EDGES;                   // row 1
    float* out = (float*)d_out;                           // [N,40] fp32

    // Workspace layout (~29.2 MB, all segments 256B-aligned): dinv | h1 | agg1 | h2
    float* ws   = (float*)d_ws;
    size_t off  = 0;
    float* dinv = ws + off; off += NODES;  off = (off + 63) & ~(size_t)63;
    float* h1   = ws + off; off += (size_t)NODES * FHID;
    float* agg1 = ws + off; off += (size_t)NODES * FHID;
    float* h2   = ws + off; off += (size_t)NODES * FOUT;
    (void)ws_size; (void)in_sizes; (void)n_in; (void)out_size;

    const int B = 256;
    const int tiles = NODES / 16;   // 6250
    const int wpb   = 4;            // waves per block (128 threads)
    const long long work = (long long)EDGES + NODES;
    const int gridE = (int)((work + B - 1) / B);

    // 1) degree: init to 1.0 (self-loop), atomic count, rsqrt -> dinv (in place)
    fill_f32<<<(NODES + B - 1) / B, B, 0, stream>>>(dinv, 1.0f, NODES);
    deg_count_kernel<<<(EDGES + B - 1) / B, B, 0, stream>>>(dstI, dinv, EDGES);
    dinv_kernel<<<(NODES + B - 1) / B, B, 0, stream>>>(dinv, NODES);

    // 2) zero accumulators (every call: graph replays must be deterministic)
    fill_f32<<<(NODES * FHID + B - 1) / B, B, 0, stream>>>(agg1, 0.0f, NODES * FHID);
    fill_f32<<<(NODES * FOUT + B - 1) / B, B, 0, stream>>>(out, 0.0f, NODES * FOUT);

    // 3) layer 1: WMMA transform, then edge scatter
    gemm1_wmma_kernel<<<(tiles + wpb - 1) / wpb, wpb * 32, 0, stream>>>(x, W1, h1);
    edge_agg_kernel<FHID><<<gridE, B, 0, stream>>>(srcI, dstI, dinv, h1, agg1);

    // 4) layer 2: fused bias+ReLU into WMMA transform, then edge scatter into out
    gemm2_wmma_kernel<<<(tiles + wpb - 1) / wpb, wpb * 32, 0, stream>>>(agg1, b1, W2, h2);
    edge_agg_kernel<FOUT><<<gridE, B, 0, stream>>>(srcI, dstI, dinv, h2, out);

    // 5) +b2 and log_softmax, in place on d_out
    finalize_kernel<<<(NODES + B - 1) / B, B, 0, stream>>>(out, b2, NODES);
}